// ResidualGraphConvolutionalNetwork_1649267442178
// MI455X (gfx1250) — compile-verified
//
#include <hip/hip_runtime.h>
#include <hip/hip_bf16.h>

// -----------------------------------------------------------------------------
// Residual GCN (2 layers, shared weights) + L2-normalize + gram decoder.
// MI455X / gfx1250: wave32; V_WMMA_F32_16X16X4_F32 for all dense GEMMs (both
// are NT-form C = A[M,K] @ B[N,K]^T, K contiguous), staged through LDS by the
// Tensor Data Mover (tensor_load_to_lds, TENSORcnt) with double buffering.
//
// Roofline: gram = 68.7 GFLOP f32 (dominant, matrix-pipe bound) + 268 MB
// output write (~11.5 us @ 23.3 TB/s). fp32 accuracy matches the reference.
// TDM pad feature (4 dwords per 32 stored dwords) gives LDS row stride 36
// dwords: 16-byte aligned rows and conflict-free (36*l mod 64 distinct for
// l=0..15) fragment reads.
// Workspace: 5 slots x N*H*4B = 80 MB.
// -----------------------------------------------------------------------------

#define N_NODES 8192
#define HDIM    512
#define NEDGE   262144
#define DECAY   0.4f
#define LSTR    36          // LDS row stride in dwords (32 data + 4 pad)

typedef __attribute__((ext_vector_type(2))) float v2f;
typedef __attribute__((ext_vector_type(8))) float v8f;

#if defined(__gfx1250__) && __has_builtin(__builtin_amdgcn_tensor_load_to_lds)
#define USE_TDM 1
#else
#define USE_TDM 0
#endif

__device__ __forceinline__ v8f wmma_f32(v2f a, v2f b, v8f c) {
    // D = A(16x4) * B(4x16) + C(16x16), full fp32, RNE.
    return __builtin_amdgcn_wmma_f32_16x16x4_f32(
        false, a, false, b, (short)0, c, false, false);
}

#if USE_TDM
typedef unsigned int uint32x4 __attribute__((ext_vector_type(4)));
typedef int          int32x4  __attribute__((ext_vector_type(4)));
typedef int          int32x8  __attribute__((ext_vector_type(8)));

__device__ __forceinline__ unsigned lds_off_of(const void* p) {
    // Low 32 bits of a generic LDS pointer are the wave-relative LDS address.
    return (unsigned)(unsigned long long)(size_t)p;
}

// Issue one TDM descriptor: tile_rows x 32 f32 tile of a row-major [rows,512]
// tensor, starting at (row0, col0), DMA'd into LDS with 4-dword padding after
// every 32 stored dwords (=> LDS row stride 36 dwords).
__device__ __forceinline__ void tdm_load_tile(const float* gbase, int row0, int col0,
                                              int tensor_rows, int tile_rows,
                                              unsigned lds_byte) {
    unsigned long long ga =
        (unsigned long long)(size_t)(gbase + (size_t)row0 * HDIM + col0);
    uint32x4 g0;
    g0[0] = 1u;                                          // count=1, user mode
    g0[1] = lds_byte;                                    // lds_addr
    g0[2] = (unsigned)(ga & 0xFFFFFFFFu);                // global_addr[31:0]
    g0[3] = (unsigned)((ga >> 32) & 0x1FFFFFFu)          // global_addr[56:32]
          | (2u << 30);                                  // type = 2 (image)
    int32x8 g1;
    g1[0] = (2 << 16)                                    // data_size = 4 bytes
          | (1 << 20)                                    // pad_enable
          | (4 << 22)                                    // pad_interval: 32 dw
          | (3 << 25);                                   // pad_amount:   4 dw
    g1[1] = (HDIM & 0xFFFF) << 16;                       // tensor_dim0 = 512
    g1[2] = (tensor_rows & 0xFFFF) << 16;                // tensor_dim1 lo
    g1[3] = ((tensor_rows >> 16) & 0xFFFF)               // tensor_dim1 hi
          | (32 << 16);                                  // tile_dim0 = 32
    g1[4] = tile_rows & 0xFFFF;                          // tile_dim1 (dim2=0)
    g1[5] = HDIM;                                        // tensor_dim0_stride
    g1[6] = 0;
    g1[7] = 0;
    int32x4 gz  = {0, 0, 0, 0};                          // dims >=2 unused
    int32x8 gz8 = {0, 0, 0, 0, 0, 0, 0, 0};
    __builtin_amdgcn_tensor_load_to_lds(g0, g1, gz, gz, gz8, 0);
}
#endif // USE_TDM

// Synchronous fallback tile loader: ROWS x 32 f32, 256 threads, float4 chunks.
template <int ROWS>
__device__ __forceinline__ void sync_load_tile(float (*dst)[LSTR],
                                               const float* __restrict__ src,
                                               int row0, int col0, int tid) {
#pragma unroll
    for (int q = tid; q < ROWS * 8; q += 256) {
        int r = q >> 3;
        int c = (q & 7) << 2;
        float4 v = *(const float4*)(src + (size_t)(row0 + r) * HDIM + col0 + c);
        dst[r][c + 0] = v.x;
        dst[r][c + 1] = v.y;
        dst[r][c + 2] = v.z;
        dst[r][c + 3] = v.w;
    }
}

// ---------------------------------------------------------------- elementwise
__global__ __launch_bounds__(256) void zero_kernel(float4* __restrict__ p, int n4) {
    int i = blockIdx.x * 256 + threadIdx.x;
    if (i < n4) p[i] = make_float4(0.f, 0.f, 0.f, 0.f);
}

__global__ __launch_bounds__(256) void mul_kernel(const float4* __restrict__ a,
                                                  const float4* __restrict__ b,
                                                  float4* __restrict__ y, int n4) {
    int i = blockIdx.x * 256 + threadIdx.x;
    if (i < n4) {
        float4 u = a[i], v = b[i];
        y[i] = make_float4(u.x * v.x, u.y * v.y, u.z * v.z, u.w * v.w);
    }
}

// -------------------------------------------------------------------- SpMM
// COO scatter-add: one wave per edge, lanes stride H. The 16 MB accumulator
// lives in the 192 MB L2, so the f32 atomics stay on-chip.
__global__ __launch_bounds__(256) void spmm_kernel(const int* __restrict__ rows,
                                                   const int* __restrict__ cols,
                                                   const float* __restrict__ vals,
                                                   const float* __restrict__ x,
                                                   float* __restrict__ out) {
    int e    = blockIdx.x * 8 + (threadIdx.x >> 5);
    int lane = threadIdx.x & 31;
    int r = rows[e], c = cols[e];
    float v = vals[e];
    const float* xr = x + (size_t)c * HDIM;
    float* orow     = out + (size_t)r * HDIM;
#pragma unroll
    for (int i = 0; i < HDIM / 32; ++i) {
        int h = lane + i * 32;
        atomicAdd(orow + h, v * xr[h]);
    }
}

// -------------------------------------------------- fused layer GEMM + epilogue
// pre  = A @ W1^T + A2 @ W2^T + b1 + b2
// act  = elu(pre);  xnew = residual ? residual + DECAY*act : act
// Block tile 64x64, 8 waves, wave = 16 rows x 32 cols (2 accumulators).
// TDM double-buffers 4 tiles/chunk; wave 0 drives the DMA.
__global__ __launch_bounds__(256) void gemm_layer_kernel(
    const float* __restrict__ A, const float* __restrict__ A2,
    const float* __restrict__ W1, const float* __restrict__ W2,
    const float* __restrict__ b1, const float* __restrict__ b2,
    const float* residual, float decay,
    float* pre_out, float* x_out) {
    __shared__ float lA[2][64][LSTR], lA2[2][64][LSTR];
    __shared__ float lB[2][64][LSTR], lB2[2][64][LSTR];

    const int tid  = threadIdx.x;
    const int n0   = blockIdx.x * 64;        // output cols == rows of W
    const int m0   = blockIdx.y * 64;        // output rows == rows of A
    const int w    = tid >> 5;
    const int lane = tid & 31;
    const int mOff = (w >> 1) * 16;
    const int nOff = (w & 1) * 32;
    const int l15  = lane & 15;
    const int hi2  = (lane >> 4) * 2;

    v8f c0 = {}, c1 = {};

#if USE_TDM
    if (w == 0) {
        tdm_load_tile(A,  m0, 0, N_NODES, 64, lds_off_of(&lA [0][0][0]));
        tdm_load_tile(A2, m0, 0, N_NODES, 64, lds_off_of(&lA2[0][0][0]));
        tdm_load_tile(W1, n0, 0, HDIM,    64, lds_off_of(&lB [0][0][0]));
        tdm_load_tile(W2, n0, 0, HDIM,    64, lds_off_of(&lB2[0][0][0]));
    }
#endif
    for (int kc = 0; kc < HDIM; kc += 32) {
        const int buf = (kc >> 5) & 1;
#if USE_TDM
        if (w == 0) {
            if (kc + 32 < HDIM) {
                tdm_load_tile(A,  m0, kc + 32, N_NODES, 64, lds_off_of(&lA [buf ^ 1][0][0]));
                tdm_load_tile(A2, m0, kc + 32, N_NODES, 64, lds_off_of(&lA2[buf ^ 1][0][0]));
                tdm_load_tile(W1, n0, kc + 32, HDIM,    64, lds_off_of(&lB [buf ^ 1][0][0]));
                tdm_load_tile(W2, n0, kc + 32, HDIM,    64, lds_off_of(&lB2[buf ^ 1][0][0]));
                __builtin_amdgcn_s_wait_tensorcnt(4);   // current buf complete
            } else {
                __builtin_amdgcn_s_wait_tensorcnt(0);
            }
        }
#else
        sync_load_tile<64>(lA [buf], A,  m0, kc, tid);
        sync_load_tile<64>(lA2[buf], A2, m0, kc, tid);
        sync_load_tile<64>(lB [buf], W1, n0, kc, tid);
        sync_load_tile<64>(lB2[buf], W2, n0, kc, tid);
#endif
        __syncthreads();
#pragma unroll
        for (int kk = 0; kk < 32; kk += 4) {
            v2f a, a2, bx, by, b2x, b2y;
            a.x   = lA [buf][mOff + l15][kk + hi2];      a.y   = lA [buf][mOff + l15][kk + hi2 + 1];
            a2.x  = lA2[buf][mOff + l15][kk + hi2];      a2.y  = lA2[buf][mOff + l15][kk + hi2 + 1];
            bx.x  = lB [buf][nOff + l15][kk + hi2];      bx.y  = lB [buf][nOff + l15][kk + hi2 + 1];
            by.x  = lB [buf][nOff + 16 + l15][kk + hi2]; by.y  = lB [buf][nOff + 16 + l15][kk + hi2 + 1];
            b2x.x = lB2[buf][nOff + l15][kk + hi2];      b2x.y = lB2[buf][nOff + l15][kk + hi2 + 1];
            b2y.x = lB2[buf][nOff + 16 + l15][kk + hi2]; b2y.y = lB2[buf][nOff + 16 + l15][kk + hi2 + 1];
            c0 = wmma_f32(a,  bx,  c0);
            c0 = wmma_f32(a2, b2x, c0);
            c1 = wmma_f32(a,  by,  c1);
            c1 = wmma_f32(a2, b2y, c1);
        }
        __syncthreads();   // all readers done before next TDM write / reload
    }

    const int rowBase = m0 + mOff + (lane >> 4) * 8;  // D layout: M=r / M=r+8
#pragma unroll
    for (int r = 0; r < 8; ++r) {
        int row = rowBase + r;
        int col = n0 + nOff + l15;
#pragma unroll
        for (int t = 0; t < 2; ++t) {
            int cc = col + t * 16;
            float pre = (t == 0 ? c0[r] : c1[r]) + b1[cc] + b2[cc];
            float act = pre > 0.f ? pre : expm1f(pre);   // elu, alpha=1
            float xn  = residual ? residual[(size_t)row * HDIM + cc] + decay * act
                                 : act;
            if (pre_out) pre_out[(size_t)row * HDIM + cc] = pre;
            x_out[(size_t)row * HDIM + cc] = xn;
        }
    }
}

// ------------------------------------------------------- row L2 normalization
__global__ __launch_bounds__(256) void normalize_kernel(const float* __restrict__ src,
                                                        float* __restrict__ hid) {
    __shared__ float red[8];
    const int row = blockIdx.x, tid = threadIdx.x;
    const float* p = src + (size_t)row * HDIM;
    float s = 0.f;
    for (int h = tid; h < HDIM; h += 256) { float v = p[h]; s += v * v; }
#pragma unroll
    for (int off = 16; off > 0; off >>= 1) s += __shfl_down(s, off, 32);
    if ((tid & 31) == 0) red[tid >> 5] = s;
    __syncthreads();
    float tot = red[0] + red[1] + red[2] + red[3] + red[4] + red[5] + red[6] + red[7];
    float norm = fmaxf(sqrtf(tot), 1e-12f);
    for (int h = tid; h < HDIM; h += 256) hid[(size_t)row * HDIM + h] = p[h] / norm;
}

// --------------------------------------------------------- gram: relu(H @ H^T)
// Dominant GEMM: block tile 64x128, 8 waves in 4x2, wave = 16 rows x 64 cols
// (4 accumulators => each A fragment feeds 4 WMMAs). TDM double-buffered.
__global__ __launch_bounds__(256) void gram_kernel(const float* __restrict__ hid,
                                                   float* __restrict__ adj) {
    __shared__ float lA[2][64][LSTR];
    __shared__ float lB[2][128][LSTR];

    const int tid  = threadIdx.x;
    const int n0   = blockIdx.x * 128;
    const int m0   = blockIdx.y * 64;
    const int w    = tid >> 5;
    const int lane = tid & 31;
    const int mOff = (w & 3) * 16;
    const int nOff = (w >> 2) * 64;
    const int l15  = lane & 15;
    const int hi2  = (lane >> 4) * 2;

    v8f c0 = {}, c1 = {}, c2 = {}, c3 = {};

#if USE_TDM
    if (w == 0) {
        tdm_load_tile(hid, m0, 0, N_NODES, 64,  lds_off_of(&lA[0][0][0]));
        tdm_load_tile(hid, n0, 0, N_NODES, 128, lds_off_of(&lB[0][0][0]));
    }
#endif
    for (int kc = 0; kc < HDIM; kc += 32) {
        const int buf = (kc >> 5) & 1;
#if USE_TDM
        if (w == 0) {
            if (kc + 32 < HDIM) {
                tdm_load_tile(hid, m0, kc + 32, N_NODES, 64,  lds_off_of(&lA[buf ^ 1][0][0]));
                tdm_load_tile(hid, n0, kc + 32, N_NODES, 128, lds_off_of(&lB[buf ^ 1][0][0]));
                __builtin_amdgcn_s_wait_tensorcnt(2);
            } else {
                __builtin_amdgcn_s_wait_tensorcnt(0);
            }
        }
#else
        sync_load_tile<64>(lA[buf],  hid, m0, kc, tid);
        sync_load_tile<128>(lB[buf], hid, n0, kc, tid);
#endif
        __syncthreads();
#pragma unroll
        for (int kk = 0; kk < 32; kk += 4) {
            v2f a, b0, b1v, b2v, b3v;
            a.x   = lA[buf][mOff + l15][kk + hi2];       a.y   = lA[buf][mOff + l15][kk + hi2 + 1];
            b0.x  = lB[buf][nOff +      l15][kk + hi2];  b0.y  = lB[buf][nOff +      l15][kk + hi2 + 1];
            b1v.x = lB[buf][nOff + 16 + l15][kk + hi2];  b1v.y = lB[buf][nOff + 16 + l15][kk + hi2 + 1];
            b2v.x = lB[buf][nOff + 32 + l15][kk + hi2];  b2v.y = lB[buf][nOff + 32 + l15][kk + hi2 + 1];
            b3v.x = lB[buf][nOff + 48 + l15][kk + hi2];  b3v.y = lB[buf][nOff + 48 + l15][kk + hi2 + 1];
            c0 = wmma_f32(a, b0,  c0);
            c1 = wmma_f32(a, b1v, c1);
            c2 = wmma_f32(a, b2v, c2);
            c3 = wmma_f32(a, b3v, c3);
        }
        __syncthreads();
    }

    const int rowBase = m0 + mOff + (lane >> 4) * 8;
#pragma unroll
    for (int r = 0; r < 8; ++r) {
        size_t row = (size_t)(rowBase + r) * N_NODES;
        int col = n0 + nOff + l15;
        adj[row + col +  0] = fmaxf(c0[r], 0.f);
        adj[row + col + 16] = fmaxf(c1[r], 0.f);
        adj[row + col + 32] = fmaxf(c2[r], 0.f);
        adj[row + col + 48] = fmaxf(c3[r], 0.f);
    }
}

// -----------------------------------------------------------------------------
extern "C" void kernel_launch(void* const* d_in, const int* in_sizes, int n_in,
                              void* d_out, int out_size, void* d_ws, size_t ws_size,
                              hipStream_t stream) {
    const float* x0   = (const float*)d_in[0];
    const int*   rows = (const int*)d_in[1];
    const int*   cols = (const int*)d_in[2];
    const float* vals = (const float*)d_in[3];
    const float* W1   = (const float*)d_in[4];
    const float* b1   = (const float*)d_in[5];
    const float* W2   = (const float*)d_in[6];
    const float* b2   = (const float*)d_in[7];

    float* out = (float*)d_out;
    float* hid = out + (size_t)N_NODES * N_NODES;   // hidden_emb slice of d_out

    const size_t SLOT = (size_t)N_NODES * HDIM;     // 4M floats = 16 MB
    float* s0 = (float*)d_ws + 0 * SLOT;            // Ax
    float* s1 = (float*)d_ws + 1 * SLOT;            // y1 -> x1
    float* s2 = (float*)d_ws + 2 * SLOT;            // Axx1 -> y2
    float* s3 = (float*)d_ws + 3 * SLOT;            // pre1 (residual) -> x2
    float* s4 = (float*)d_ws + 4 * SLOT;            // Axx2

    const int n4       = (int)(SLOT / 4);
    const int ewBlocks = NEDGE / 8;
    dim3 gemmGridL(HDIM / 64, N_NODES / 64);        // 8 x 128
    dim3 gramGrid(N_NODES / 128, N_NODES / 64);     // 64 x 128

    // ---------------- layer 1 (input x0) ----------------
    zero_kernel<<<n4 / 256, 256, 0, stream>>>((float4*)s0, n4);
    spmm_kernel<<<ewBlocks, 256, 0, stream>>>(rows, cols, vals, x0, s0);        // Ax1
    mul_kernel<<<n4 / 256, 256, 0, stream>>>((const float4*)s0, (const float4*)x0,
                                             (float4*)s1, n4);                  // Ax1*x0
    zero_kernel<<<n4 / 256, 256, 0, stream>>>((float4*)s2, n4);
    spmm_kernel<<<ewBlocks, 256, 0, stream>>>(rows, cols, vals, s1, s2);        // Axx1
    gemm_layer_kernel<<<gemmGridL, 256, 0, stream>>>(s0, s2, W1, W2, b1, b2,
                                                     nullptr, DECAY, s3, s1);   // pre1->s3, x1->s1

    // ---------------- layer 2 (input s1, residual s3) ----------------
    zero_kernel<<<n4 / 256, 256, 0, stream>>>((float4*)s0, n4);
    spmm_kernel<<<ewBlocks, 256, 0, stream>>>(rows, cols, vals, s1, s0);        // Ax2
    mul_kernel<<<n4 / 256, 256, 0, stream>>>((const float4*)s0, (const float4*)s1,
                                             (float4*)s2, n4);                  // Ax2*x1
    zero_kernel<<<n4 / 256, 256, 0, stream>>>((float4*)s4, n4);
    spmm_kernel<<<ewBlocks, 256, 0, stream>>>(rows, cols, vals, s2, s4);        // Axx2
    // x2 = pre1 + 0.4*elu(pre2); x_out aliases residual element-for-element (safe)
    gemm_layer_kernel<<<gemmGridL, 256, 0, stream>>>(s0, s4, W1, W2, b1, b2,
                                                     s3, DECAY, nullptr, s3);

    // ---------------- decoder ----------------
    normalize_kernel<<<N_NODES, 256, 0, stream>>>(s3, hid);
    gram_kernel<<<gramGrid, 256, 0, stream>>>(hid, out);
}